// AttentionModel_87325275062870
// MI455X (gfx1250) — compile-verified
//
#include <hip/hip_runtime.h>

typedef float v2f __attribute__((ext_vector_type(2)));
typedef float v4f __attribute__((ext_vector_type(4)));
typedef float v8f __attribute__((ext_vector_type(8)));

// Problem constants (match reference)
#define BB   128
#define NN   250
#define DD   128
#define HH   8
#define FF   512
#define LL   2
#define DKk  16
#define NT   251                      // NTOT = N + 1
#define MROW (BB * NT)                // 32128 = 502 * 64
#define NEGV (-1e9f)

static __device__ __forceinline__ v8f wmma_f32(v2f a, v2f b, v8f c) {
  // V_WMMA_F32_16X16X4_F32 : D = A(16x4,f32) * B(4x16,f32) + C(16x16,f32)
  return __builtin_amdgcn_wmma_f32_16x16x4_f32(false, a, false, b, (short)0, c,
                                               false, false);
}

// Async DMA: 16 bytes global -> LDS per enabled lane (ASYNCcnt-tracked).
// LDS operand is the wave-relative byte offset = low 32 bits of generic addr.
static __device__ __forceinline__ void async_copy16(const float* g, float* l) {
  unsigned loff = (unsigned)(size_t)l;
  asm volatile("global_load_async_to_lds_b128 %0, %1, off" ::"v"(loff),
               "v"((const void*)g)
               : "memory");
}
static __device__ __forceinline__ void async_wait0() {
  asm volatile("s_wait_asynccnt 0x0" ::: "memory");
}

// ---------------------------------------------------------------------------
// fp32 WMMA GEMM, 64x64 block per wave (16 accumulators, 16 WMMAs / K-step):
//   C[M,N] = A[M,K] @ W[K,N] (+bias)(+relu)
// M and N must be multiples of 64 (true for all call sites here).
// EPI: 0 = none, 1 = +bias, 2 = +bias+relu
// ---------------------------------------------------------------------------
template <int EPI>
__global__ __launch_bounds__(256) void gemm_wmma_k(
    const float* __restrict__ A, const float* __restrict__ W,
    const float* __restrict__ bias, float* __restrict__ C, int Mr, int Nr,
    int Kr) {
  int gtid = blockIdx.x * 256 + threadIdx.x;
  int wave = gtid >> 5;
  int lane = gtid & 31;
  int tnc = Nr >> 6;                       // 64-wide column blocks
  int tm = wave / tnc, tn = wave % tnc;
  if (tm * 64 >= Mr) return;               // wave-uniform guard (EXEC full)
  int l16 = lane & 15, half = lane >> 4;
  int row0 = tm * 64 + l16;
  int col0 = tn * 64 + l16;

  v8f acc[4][4];
#pragma unroll
  for (int mi = 0; mi < 4; ++mi)
#pragma unroll
    for (int ni = 0; ni < 4; ++ni) acc[mi][ni] = (v8f){};

  for (int k0 = 0; k0 < Kr; k0 += 4) {
    int ka = k0 + half * 2;
    if (k0 + 8 < Kr)  // prefetch next B panel rows (global_prefetch_b8)
      __builtin_prefetch(W + (size_t)(ka + 8) * Nr + col0, 0, 1);
    v2f a[4], b[4];
#pragma unroll
    for (int mi = 0; mi < 4; ++mi)
      a[mi] = *(const v2f*)(A + (size_t)(row0 + mi * 16) * Kr + ka);
#pragma unroll
    for (int ni = 0; ni < 4; ++ni) {
      b[ni].x = W[(size_t)ka * Nr + col0 + ni * 16];
      b[ni].y = W[(size_t)(ka + 1) * Nr + col0 + ni * 16];
    }
#pragma unroll
    for (int mi = 0; mi < 4; ++mi)
#pragma unroll
      for (int ni = 0; ni < 4; ++ni)
        acc[mi][ni] = wmma_f32(a[mi], b[ni], acc[mi][ni]);
  }
#pragma unroll
  for (int ni = 0; ni < 4; ++ni) {
    int cb = col0 + ni * 16;
    float bv = (EPI >= 1) ? bias[cb] : 0.f;
#pragma unroll
    for (int mi = 0; mi < 4; ++mi) {
#pragma unroll
      for (int v = 0; v < 8; ++v) {
        int r = tm * 64 + mi * 16 + v + half * 8;
        float vv = acc[mi][ni][v] + bv;
        if (EPI == 2) vv = fmaxf(vv, 0.f);
        C[(size_t)r * Nr + cb] = vv;
      }
    }
  }
}

// ---------------------------------------------------------------------------
// Init embed: h[b,0,:] = depot@idw+idb ; h[b,1+n,:] = [coords,demand]@inw+inb
// ---------------------------------------------------------------------------
__global__ __launch_bounds__(256) void init_embed_k(
    const float* __restrict__ depot, const float* __restrict__ coords,
    const float* __restrict__ demand, const float* __restrict__ idw,
    const float* __restrict__ idb, const float* __restrict__ inw,
    const float* __restrict__ inb, float* __restrict__ h) {
  int i = blockIdx.x * 256 + threadIdx.x;
  if (i >= MROW * DD) return;
  int d = i & 127;
  int row = i >> 7;
  int b = row / NT, n = row % NT;
  float v;
  if (n == 0) {
    v = depot[b * 2] * idw[d] + depot[b * 2 + 1] * idw[DD + d] + idb[d];
  } else {
    int c = n - 1;
    v = coords[(b * NN + c) * 2] * inw[d] +
        coords[(b * NN + c) * 2 + 1] * inw[DD + d] +
        demand[b * NN + c] * inw[2 * DD + d] + inb[d];
  }
  h[i] = v;
}

// ---------------------------------------------------------------------------
// Repack Wq|Wk|Wv: [L,H,D,DK]x3 -> one [L, D, 384] weight (QKV in one GEMM)
// ---------------------------------------------------------------------------
__global__ __launch_bounds__(256) void repack_qkv_k(
    const float* __restrict__ Wq, const float* __restrict__ Wk,
    const float* __restrict__ Wv, float* __restrict__ pqkv) {
  int i = blockIdx.x * 256 + threadIdx.x;
  if (i >= LL * DD * 384) return;
  int c = i % 384;
  int d = (i / 384) & 127;
  int l = i / (384 * DD);
  const float* Wsrc = (c < 128) ? Wq : (c < 256) ? Wk : Wv;
  int hk = c & 127;
  int src = ((l * HH + (hk >> 4)) * DD + d) * DKk + (hk & 15);
  pqkv[i] = Wsrc[src];
}

// ---------------------------------------------------------------------------
// Encoder MHA core per (head,batch): S = QK^T/4, softmax, O = S@V via WMMA.
// QKV rows DMA'd into LDS with global_load_async_to_lds_b128 (ASYNCcnt),
// zero-padded to 256 rows; A-fragments hoisted out of the column-tile loop.
// QKV layout: [M, 384] = [q(128) | k(128) | v(128)] per row.
// ---------------------------------------------------------------------------
__global__ __launch_bounds__(256) void attn_k(const float* __restrict__ QKV,
                                              float* __restrict__ Oc) {
  extern __shared__ float sm[];
  float* Qs = sm;                  // 256*16
  float* Ks = Qs + 256 * DKk;      // 256*16
  float* Vs = Ks + 256 * DKk;      // 256*16
  float* Sb = Vs + 256 * DKk;      // 8 * 16 * 256
  int hb = blockIdx.x;
  int h = hb / BB;
  int b = hb % BB;
  int tid = threadIdx.x;
  int w = tid >> 5, lane = tid & 31;
  int l16 = lane & 15, half = lane >> 4;
  // stage Q,K,V rows: async DMA for valid rows, zero-fill for pad rows
  if (tid < NT) {
    const float* base = QKV + (size_t)(b * NT + tid) * 384 + h * DKk;
#pragma unroll
    for (int k4 = 0; k4 < 4; ++k4) {
      async_copy16(base + k4 * 4, Qs + tid * DKk + k4 * 4);
      async_copy16(base + 128 + k4 * 4, Ks + tid * DKk + k4 * 4);
      async_copy16(base + 256 + k4 * 4, Vs + tid * DKk + k4 * 4);
    }
  } else {
#pragma unroll
    for (int k = 0; k < DKk; ++k) {
      Qs[tid * DKk + k] = 0.f;
      Ks[tid * DKk + k] = 0.f;
      Vs[tid * DKk + k] = 0.f;
    }
  }
  async_wait0();
  __syncthreads();
  float* S = Sb + w * 16 * 256;
  for (int rtb = 0; rtb < 2; ++rtb) {
    int rt = w + rtb * 8;  // row tile 0..15
    // hoisted A fragments for this row tile (from LDS, b64 loads)
    v2f afrag[4];
#pragma unroll
    for (int k0 = 0; k0 < DKk; k0 += 4)
      afrag[k0 >> 2] = *(const v2f*)(Qs + (rt * 16 + l16) * DKk + k0 + half * 2);
    // ---- scores: S[16,256] = Q_tile @ K^T * 0.25 (pad cols handled later) --
    for (int ct = 0; ct < 16; ++ct) {
      v8f acc = {};
      int kn = ct * 16 + l16;
#pragma unroll
      for (int k0 = 0; k0 < DKk; k0 += 4) {
        v2f bb = *(const v2f*)(Ks + kn * DKk + k0 + half * 2);
        acc = wmma_f32(afrag[k0 >> 2], bb, acc);
      }
      int col = ct * 16 + l16;
#pragma unroll
      for (int v = 0; v < 8; ++v) S[(v + half * 8) * 256 + col] = acc[v] * 0.25f;
    }
    __syncthreads();
    // ---- row softmax over n<NT (lane<16 of each wave owns one row);
    //      pad columns NT..255 zeroed so the AV WMMA ignores them ----
    if (lane < 16) {
      float* srow = S + lane * 256;
      if (rt * 16 + lane < NT) {
        float m = -1e30f;
        for (int n = 0; n < NT; ++n) m = fmaxf(m, srow[n]);
        float s = 0.f;
        for (int n = 0; n < NT; ++n) {
          float e = __expf(srow[n] - m);
          srow[n] = e;
          s += e;
        }
        float inv = 1.f / s;
        for (int n = 0; n < NT; ++n) srow[n] *= inv;
        for (int n = NT; n < 256; ++n) srow[n] = 0.f;
      } else {
        for (int n = 0; n < 256; ++n) srow[n] = 0.f;
      }
    }
    __syncthreads();
    // ---- O_tile[16,16] = P[16,256] @ V[256,16] ----
    v8f acc = {};
    for (int k0 = 0; k0 < 256; k0 += 4) {
      int ka = k0 + half * 2;
      v2f a = *(const v2f*)(S + l16 * 256 + ka);
      v2f bb;
      bb.x = Vs[ka * DKk + l16];
      bb.y = Vs[(ka + 1) * DKk + l16];
      acc = wmma_f32(a, bb, acc);
    }
#pragma unroll
    for (int v = 0; v < 8; ++v) {
      int r = rt * 16 + v + half * 8;
      if (r < NT) Oc[(size_t)(b * NT + r) * DD + h * DKk + l16] = acc[v];
    }
    __syncthreads();
  }
}

// ---------------------------------------------------------------------------
// BatchNorm (training-mode batch stats, biased var) on (x + r)
// ---------------------------------------------------------------------------
__global__ __launch_bounds__(256) void bn_stats_k(const float* __restrict__ x,
                                                  const float* __restrict__ r,
                                                  float* __restrict__ mean,
                                                  float* __restrict__ rstd) {
  int d = blockIdx.x;
  int tid = threadIdx.x;
  __shared__ float s1[256], s2[256];
  float a = 0.f, q = 0.f;
  for (int i = tid; i < MROW; i += 256) {
    float v = x[(size_t)i * DD + d] + r[(size_t)i * DD + d];
    a += v;
    q += v * v;
  }
  s1[tid] = a;
  s2[tid] = q;
  __syncthreads();
  for (int st = 128; st > 0; st >>= 1) {
    if (tid < st) {
      s1[tid] += s1[tid + st];
      s2[tid] += s2[tid + st];
    }
    __syncthreads();
  }
  if (tid == 0) {
    float m = s1[0] / (float)MROW;
    float v = s2[0] / (float)MROW - m * m;
    mean[d] = m;
    rstd[d] = rsqrtf(v + 1e-5f);
  }
}

__global__ __launch_bounds__(256) void bn_apply_k(
    const float* __restrict__ x, const float* __restrict__ r,
    const float* __restrict__ mean, const float* __restrict__ rstd,
    const float* __restrict__ g, const float* __restrict__ bta,
    float* __restrict__ y) {
  int i = blockIdx.x * 256 + threadIdx.x;
  if (i >= MROW * DD) return;
  int d = i & 127;
  float v = x[i] + r[i];
  y[i] = (v - mean[d]) * rstd[d] * g[d] + bta[d];
}

// mean over nodes: memb[b,d] = mean_n emb[b,n,d]
__global__ __launch_bounds__(256) void row_mean_k(const float* __restrict__ emb,
                                                  float* __restrict__ mo) {
  int i = blockIdx.x * 256 + threadIdx.x;
  if (i >= BB * DD) return;
  int d = i & 127, b = i >> 7;
  float s = 0.f;
  for (int n = 0; n < NT; ++n) s += emb[(size_t)(b * NT + n) * DD + d];
  mo[i] = s * (1.f / (float)NT);
}

// ---------------------------------------------------------------------------
// Greedy pointer decoder: one workgroup per batch element, 250 serial steps.
// psW/poW DMA'd once into LDS via async-to-LDS (amortized over all steps);
// emb[cur] staged per step; gk/lk dot products use b128 loads.
// out[0..B) = ll ; out[B + b*N + t] = (float)pi[b,t]
// ---------------------------------------------------------------------------
__global__ __launch_bounds__(256) void decoder_k(
    const float* __restrict__ emb, const float* __restrict__ proj,
    const float* __restrict__ fctx, const float* __restrict__ demand,
    const float* __restrict__ psW, const float* __restrict__ poW,
    float* __restrict__ out) {
  extern __shared__ float dyn[];
  float* psL = dyn;                       // 129*128
  float* poL = psL + (DD + 1) * DD;       // 128*128
  int b = blockIdx.x;
  int tid = threadIdx.x;
  int w = tid >> 5, lane = tid & 31;
  __shared__ float q[DD], gbuf[DD], fq[DD], ceS[DD];
  __shared__ float comp[HH * NT];
  __shared__ float logitsS[256];
  __shared__ float redv[256];
  __shared__ int redi[256];
  __shared__ int vis[NT];
  __shared__ float hred[HH * 32], hmax[HH], hsum[HH];
  __shared__ int curS;
  __shared__ float capS, llS;

  // one-time LDS staging of the two decoder weight matrices (async DMA)
  for (int i = tid * 4; i < (DD + 1) * DD; i += 256 * 4)
    async_copy16(psW + i, psL + i);
  for (int i = tid * 4; i < DD * DD; i += 256 * 4)
    async_copy16(poW + i, poL + i);
  for (int i = tid; i < NT; i += 256) vis[i] = (i == 0) ? 1 : 0;
  if (tid == 0) {
    curS = 0;
    capS = 1.f;
    llS = 0.f;
  }
  async_wait0();
  __syncthreads();

  for (int t = 0; t < NN; ++t) {
    int cur = curS;
    float cap = capS;
    if (tid < DD) ceS[tid] = emb[(size_t)(b * NT + cur) * DD + tid];
    __syncthreads();
    // q = fixed_ctx + [emb[cur], cap] @ psW    (psW: [D+1, D])
    if (tid < DD) {
      float s = fctx[b * DD + tid];
      for (int j = 0; j < DD; ++j) s += ceS[j] * psL[j * DD + tid];
      s += cap * psL[DD * DD + tid];
      q[tid] = s;
    }
    __syncthreads();
    // ---- glimpse: wave w == head w ----
    {
      int hh = w;
      const float* qh = q + hh * DKk;
      float mloc = -1e30f;
      for (int n = lane; n < NT; n += 32) {
        const v4f* kr4 = (const v4f*)(proj + (size_t)(b * NT + n) * 384 +
                                      hh * DKk);  // gk, 64B-aligned rows
        float s = 0.f;
#pragma unroll
        for (int k4 = 0; k4 < 4; ++k4) {
          v4f kv = kr4[k4];
          s += qh[k4 * 4 + 0] * kv.x + qh[k4 * 4 + 1] * kv.y +
               qh[k4 * 4 + 2] * kv.z + qh[k4 * 4 + 3] * kv.w;
        }
        s *= 0.25f;
        if (vis[n]) s = NEGV;
        comp[hh * NT + n] = s;
        mloc = fmaxf(mloc, s);
      }
      hred[hh * 32 + lane] = mloc;
      __syncthreads();
      if (lane == 0) {
        float m = hred[hh * 32];
        for (int j = 1; j < 32; ++j) m = fmaxf(m, hred[hh * 32 + j]);
        hmax[hh] = m;
      }
      __syncthreads();
      float m = hmax[hh];
      float sloc = 0.f;
      for (int n = lane; n < NT; n += 32) {
        float e = __expf(comp[hh * NT + n] - m);
        comp[hh * NT + n] = e;
        sloc += e;
      }
      hred[hh * 32 + lane] = sloc;
      __syncthreads();
      if (lane == 0) {
        float s = 0.f;
        for (int j = 0; j < 32; ++j) s += hred[hh * 32 + j];
        hsum[hh] = s;
      }
      __syncthreads();
      if (lane < DKk) {
        float inv = 1.f / hsum[hh];
        float acc = 0.f;
        for (int n = 0; n < NT; ++n)
          acc += comp[hh * NT + n] *
                 proj[(size_t)(b * NT + n) * 384 + 128 + hh * DKk + lane];  // gv
        gbuf[hh * DKk + lane] = acc * inv;
      }
    }
    __syncthreads();
    // ---- project_out: fq = g @ poW ----
    if (tid < DD) {
      float s = 0.f;
      for (int j = 0; j < DD; ++j) s += gbuf[j] * poL[j * DD + tid];
      fq[tid] = s;
    }
    __syncthreads();
    // ---- logits = 10*tanh(fq . lk / sqrt(D)), masked ----
    {
      float v = NEGV;
      if (tid < NT && !vis[tid]) {
        const v4f* lk4 =
            (const v4f*)(proj + (size_t)(b * NT + tid) * 384 + 256);  // lk
        float s = 0.f;
#pragma unroll 8
        for (int j4 = 0; j4 < 32; ++j4) {
          v4f lv = lk4[j4];
          s += fq[j4 * 4 + 0] * lv.x + fq[j4 * 4 + 1] * lv.y +
               fq[j4 * 4 + 2] * lv.z + fq[j4 * 4 + 3] * lv.w;
        }
        v = 10.f * tanhf(s * 0.08838834764831845f);  // 1/sqrt(128)
      }
      logitsS[tid] = v;
      redv[tid] = v;
      redi[tid] = (tid < NT) ? tid : 0x7fffffff;
    }
    __syncthreads();
    // argmax (first-index tie break, matches jnp.argmax)
    for (int st = 128; st > 0; st >>= 1) {
      if (tid < st) {
        float v2 = redv[tid + st];
        int i2 = redi[tid + st];
        if (v2 > redv[tid] || (v2 == redv[tid] && i2 < redi[tid])) {
          redv[tid] = v2;
          redi[tid] = i2;
        }
      }
      __syncthreads();
    }
    int sel = redi[0];
    float m = redv[0];
    __syncthreads();
    // logsumexp
    redv[tid] = __expf(logitsS[tid] - m);
    __syncthreads();
    for (int st = 128; st > 0; st >>= 1) {
      if (tid < st) redv[tid] += redv[tid + st];
      __syncthreads();
    }
    if (tid == 0) {
      float lse = m + __logf(redv[0]);
      llS += logitsS[sel] - lse;
      out[BB + b * NN + t] = (float)sel;
      vis[sel] = 1;
      float dd = (sel == 0) ? 0.f : demand[b * NN + sel - 1];
      capS = (sel == 0) ? 1.f : capS - dd;
      curS = sel;
    }
    __syncthreads();
  }
  if (tid == 0) out[b] = llS;
}

// ---------------------------------------------------------------------------
static inline int gemm_blocks(int Mr, int Nr) {
  long waves = (long)(Mr / 64) * (Nr / 64);
  return (int)((waves * 32 + 255) / 256);
}

extern "C" void kernel_launch(void* const* d_in, const int* in_sizes, int n_in,
                              void* d_out, int out_size, void* d_ws,
                              size_t ws_size, hipStream_t stream) {
  (void)in_sizes; (void)n_in; (void)out_size; (void)ws_size;
  const float* depot  = (const float*)d_in[0];
  const float* coords = (const float*)d_in[1];
  const float* demand = (const float*)d_in[2];
  const float* idw = (const float*)d_in[3];
  const float* idb = (const float*)d_in[4];
  const float* inw = (const float*)d_in[5];
  const float* inb = (const float*)d_in[6];
  const float* Wq = (const float*)d_in[7];
  const float* Wk = (const float*)d_in[8];
  const float* Wv = (const float*)d_in[9];
  const float* Wo = (const float*)d_in[10];
  const float* bn1g = (const float*)d_in[11];
  const float* bn1b = (const float*)d_in[12];
  const float* fW1 = (const float*)d_in[13];
  const float* fb1 = (const float*)d_in[14];
  const float* fW2 = (const float*)d_in[15];
  const float* fb2 = (const float*)d_in[16];
  const float* bn2g = (const float*)d_in[17];
  const float* bn2b = (const float*)d_in[18];
  const float* pnW = (const float*)d_in[19];
  const float* pfW = (const float*)d_in[20];
  const float* psW = (const float*)d_in[21];
  const float* poW = (const float*)d_in[22];
  float* out = (float*)d_out;

  const size_t M = MROW;
  float* ws   = (float*)d_ws;
  float* emb  = ws;                    // M*128
  float* buf1 = emb + M * DD;          // M*512 (QKV(384)|Hc(128), later ff1/proj)
  float* buf2 = buf1 + M * (size_t)FF; // M*128 (attn-out / ff2-out)
  float* pqkv = buf2 + M * DD;         // 2*128*384
  float* mnb  = pqkv + LL * DD * 384;  // 128
  float* rsd  = mnb + DD;              // 128
  float* memb = rsd + DD;              // 128*128
  float* fctx = memb + BB * DD;        // 128*128

  float* QKVb = buf1;                  // M*384
  float* Hc   = buf1 + 3 * M * DD;     // M*128

  const int eltBlocks = (int)((M * DD + 255) / 256);

  repack_qkv_k<<<(LL * DD * 384 + 255) / 256, 256, 0, stream>>>(Wq, Wk, Wv,
                                                                pqkv);
  init_embed_k<<<eltBlocks, 256, 0, stream>>>(depot, coords, demand, idw, idb,
                                              inw, inb, emb);

  const size_t attn_smem = (3 * 256 * DKk + 8 * 16 * 256) * sizeof(float);
  const size_t dec_smem  = ((DD + 1) * DD + DD * DD) * sizeof(float);

  for (int l = 0; l < LL; ++l) {
    // fused Q|K|V projection (one fp32 WMMA GEMM, N=384)
    gemm_wmma_k<0><<<gemm_blocks(M, 384), 256, 0, stream>>>(
        emb, pqkv + (size_t)l * DD * 384, nullptr, QKVb, M, 384, DD);
    // attention per (head,batch)
    attn_k<<<HH * BB, 256, attn_smem, stream>>>(QKVb, Hc);
    // out-projection (Wo[l] already [H*DK, D] row-major)
    gemm_wmma_k<0><<<gemm_blocks(M, DD), 256, 0, stream>>>(
        Hc, Wo + (size_t)l * DD * DD, nullptr, buf2, M, DD, DD);
    // BN(h + attn)
    bn_stats_k<<<DD, 256, 0, stream>>>(emb, buf2, mnb, rsd);
    bn_apply_k<<<eltBlocks, 256, 0, stream>>>(emb, buf2, mnb, rsd,
                                              bn1g + l * DD, bn1b + l * DD,
                                              emb);
    // FF: relu(h@fW1+b1)@fW2+b2
    gemm_wmma_k<2><<<gemm_blocks(M, FF), 256, 0, stream>>>(
        emb, fW1 + (size_t)l * DD * FF, fb1 + l * FF, buf1, M, FF, DD);
    gemm_wmma_k<1><<<gemm_blocks(M, DD), 256, 0, stream>>>(
        buf1, fW2 + (size_t)l * FF * DD, fb2 + l * DD, buf2, M, DD, FF);
    // BN(h + ff)
    bn_stats_k<<<DD, 256, 0, stream>>>(emb, buf2, mnb, rsd);
    bn_apply_k<<<eltBlocks, 256, 0, stream>>>(emb, buf2, mnb, rsd,
                                              bn2g + l * DD, bn2b + l * DD,
                                              emb);
  }

  // precompute: fixed_ctx = mean(emb)@pfW ; proj = emb@pnW -> [gk|gv|lk]
  row_mean_k<<<(BB * DD + 255) / 256, 256, 0, stream>>>(emb, memb);
  gemm_wmma_k<0><<<gemm_blocks(BB, DD), 256, 0, stream>>>(memb, pfW, nullptr,
                                                          fctx, BB, DD, DD);
  gemm_wmma_k<0><<<gemm_blocks(M, 3 * DD), 256, 0, stream>>>(
      emb, pnW, nullptr, buf1, M, 3 * DD, DD);

  // sequential greedy decoder, one workgroup per batch element
  decoder_k<<<BB, 256, dec_smem, stream>>>(emb, buf1, fctx, demand, psW, poW,
                                           out);
}